// DiffEqLayer_67903432949824
// MI455X (gfx1250) — compile-verified
//
#include <hip/hip_runtime.h>
#include <stdint.h>

typedef __attribute__((ext_vector_type(2))) float v2f;
typedef __attribute__((ext_vector_type(8))) float v8f;

#define NT 1020
#define NB 2048
#define LSTR 20   // padded LDS row stride (floats): 80B rows, 16B aligned, spreads banks

static __device__ __forceinline__ v8f wmma4(v2f a, v2f b, v8f c) {
  // D = A(16x4 f32) * B(4x16 f32) + C(16x16 f32)
  return __builtin_amdgcn_wmma_f32_16x16x4_f32(false, a, false, b, (short)0, c, false, false);
}
static __device__ __forceinline__ float bperm_f(float v, int srcLane) {
  return __int_as_float(__builtin_amdgcn_ds_bpermute(srcLane << 2, __float_as_int(v)));
}
static __device__ __forceinline__ float sigm(float x) {
  return __builtin_amdgcn_rcpf(1.0f + __builtin_amdgcn_exp2f(-1.44269504f * x));
}
static __device__ __forceinline__ float tanh_fast(float x) {
  float e = __builtin_amdgcn_exp2f(2.88539008f * x);
  return (e - 1.0f) * __builtin_amdgcn_rcpf(e + 1.0f);
}
static __device__ __forceinline__ float clip15(float x) {
  return fminf(fmaxf(x, -1.0f), 5.0f);
}

__global__ __launch_bounds__(256) void diffeq_wmma_kernel(
    const float* __restrict__ inp,   // (NT, NB, 8)
    const float* __restrict__ state, // (NB, 4)
    const float* __restrict__ caps,  // (4)
    const float* __restrict__ Wc,    // (12, 15)
    const float* __restrict__ bc,    // (15)
    const float* __restrict__ W1,    // (12, 16)
    const float* __restrict__ b1,    // (16)
    const float* __restrict__ W2,    // (16, 4)
    const float* __restrict__ b2,    // (4)
    float* __restrict__ out)         // NT*NB*4 outputs, then NB*4 final state
{
  // 3 per-wave LDS tiles of 16 rows x LSTR floats (no cross-wave sharing -> no barriers)
  __shared__ float smem[8][3 * 16 * LSTR];
  const int tid  = threadIdx.x;
  const int lane = tid & 31;
  const int wave = tid >> 5;
  const int row  = lane & 15;  // batch row within tile (also N index for B frags)
  const int half = lane >> 4;  // selects K {0,1} vs {2,3} within a K=4 fragment
  const bool hi  = (half != 0);
  const int b    = (blockIdx.x * 8 + wave) * 16 + row;

  float* cT = &smem[wave][0];             // conducts transpose tile
  float* hT = &smem[wave][16 * LSTR];     // hidden transpose tile
  float* pT = &smem[wave][32 * LSTR];     // ploss transpose tile

  // ---- loop-invariant B fragments (weights) ----
  v2f Bc[3], Bh[3], Bp[4];
#pragma unroll
  for (int c = 0; c < 3; ++c) {
    int k = 4 * c + half * 2;
    float wc0 = (row < 15) ? Wc[k * 15 + row] : 0.0f;
    float wc1 = (row < 15) ? Wc[(k + 1) * 15 + row] : 0.0f;
    Bc[c].x = wc0; Bc[c].y = wc1;
    Bh[c].x = W1[k * 16 + row];
    Bh[c].y = W1[(k + 1) * 16 + row];
  }
#pragma unroll
  for (int c = 0; c < 4; ++c) {
    int k = 4 * c + half * 2;
    Bp[c].x = (row < 4) ? W2[k * 4 + row] : 0.0f;
    Bp[c].y = (row < 4) ? W2[(k + 1) * 4 + row] : 0.0f;
  }
  // bias-seeded accumulators (D-layout: bias depends only on N = lane&15)
  float bcv = (row < 15) ? bc[row] : 0.0f;
  float b1v = b1[row];
  float b2v = (row < 4) ? b2[row] : 0.0f;
  v8f Cc0, Ch0, Cp0;
#pragma unroll
  for (int e = 0; e < 8; ++e) { Cc0[e] = bcv; Ch0[e] = b1v; Cp0[e] = b2v; }

  // k = 0.5 * exp(caps)
  const float LOG2E = 1.44269504f;
  float4 cpv = *(const float4*)caps;
  float k0 = 0.5f * __builtin_amdgcn_exp2f(LOG2E * cpv.x);
  float k1 = 0.5f * __builtin_amdgcn_exp2f(LOG2E * cpv.y);
  float k2 = 0.5f * __builtin_amdgcn_exp2f(LOG2E * cpv.z);
  float k3 = 0.5f * __builtin_amdgcn_exp2f(LOG2E * cpv.w);

  // initial state: ALL 32 lanes hold the state of batch row `row` (halves identical)
  float4 pv = *(const float4*)(state + (size_t)b * 4);
  float p0 = pv.x, p1 = pv.y, p2 = pv.z, p3 = pv.w;

  // adj_mat is compile-time constant (from np.triu_indices(6,1) construction)
  constexpr int ADJ[4][6] = {
      {0, 0, 1, 2, 3, 4},
      {0, 0, 5, 6, 7, 8},
      {1, 5, 0, 9, 10, 11},
      {2, 6, 9, 0, 12, 13}};

  // Addressing: uniform scalar element offset (advanced per step with SALU) +
  // loop-invariant per-lane 32-bit element offset -> GVS (saddr + voffset) form.
  const int lvoff = b * 8 + half * 2;  // inp lane offset (elements)
  const int ovoff = b * 4 + half * 2;  // out lane offset (elements)
  size_t toff  = 0;                    // inp scalar offset (elements), += NB*8
  size_t otoff = 0;                    // out scalar offset (elements), += NB*4

  v2f A0 = __builtin_nontemporal_load((const v2f*)(inp + toff + lvoff));
  v2f A1 = __builtin_nontemporal_load((const v2f*)(inp + toff + lvoff + 4));

  for (int t = 0; t < NT; ++t) {
    // branchless scalar prefetch offset (re-reads current row on last iter)
    size_t toffn = (t + 1 < NT) ? (toff + (size_t)NB * 8) : toff;
    v2f A0n = __builtin_nontemporal_load((const v2f*)(inp + toffn + lvoff));
    v2f A1n = __builtin_nontemporal_load((const v2f*)(inp + toffn + lvoff + 4));

    // broadcast inp[row,1] and inp[row,6] to BOTH halves (full-EXEC bpermutes)
    float inp1v = bperm_f(A0.y, row);        // from lane row    (half 0, K=1)
    float inp6v = bperm_f(A1.x, row + 16);   // from lane row+16 (half 1, K=6)

    // sub K 8..11 = prev state; same value doubles as the outputs[t] store payload
    v2f sel;
    sel.x = hi ? p2 : p0;
    sel.y = hi ? p3 : p1;

    // outputs[t] = prev (pre-update): full-EXEC b64, halves tile the float4
    __builtin_nontemporal_store(sel, (v2f*)(out + otoff + ovoff));

    // conducts = sigmoid(sub@Wc + bc) ; h = tanh(sub@W1 + b1)
    v8f Dc = Cc0;
    Dc = wmma4(A0, Bc[0], Dc);
    Dc = wmma4(A1, Bc[1], Dc);
    Dc = wmma4(sel, Bc[2], Dc);
    v8f Dh = Ch0;
    Dh = wmma4(A0, Bh[0], Dh);
    Dh = wmma4(A1, Bh[1], Dh);
    Dh = wmma4(sel, Bh[2], Dh);
#pragma unroll
    for (int e = 0; e < 8; ++e) Dc[e] = sigm(Dc[e]);
#pragma unroll
    for (int e = 0; e < 8; ++e) Dh[e] = tanh_fast(Dh[e]);

    // D-layout -> LDS: element (M = e + 8*half, N = row)
    const int sb = (half * 8) * LSTR + row;
#pragma unroll
    for (int e = 0; e < 8; ++e) {
      cT[sb + e * LSTR] = Dc[e];
      hT[sb + e * LSTR] = Dh[e];
    }

    // ploss = |h @ W2 + b2| : reload h as A fragments (K=16 -> 4 chunks)
    v8f Dp = Cp0;
    const float* hr = hT + row * LSTR + half * 2;
#pragma unroll
    for (int c = 0; c < 4; ++c) {
      v2f Ah = *(const v2f*)(hr + 4 * c);
      Dp = wmma4(Ah, Bp[c], Dp);
    }
#pragma unroll
    for (int e = 0; e < 8; ++e) pT[sb + e * LSTR] = Dp[e];

    // back to lane=batch layout (identical addresses across halves -> broadcast)
    float4 pl4 = *(const float4*)(pT + row * LSTR);
    float4 q0 = *(const float4*)(cT + row * LSTR + 0);
    float4 q1 = *(const float4*)(cT + row * LSTR + 4);
    float4 q2 = *(const float4*)(cT + row * LSTR + 8);
    float4 q3 = *(const float4*)(cT + row * LSTR + 12);
    float cond[16] = {q0.x, q0.y, q0.z, q0.w, q1.x, q1.y, q1.z, q1.w,
                      q2.x, q2.y, q2.z, q2.w, q3.x, q3.y, q3.z, q3.w};

    float temps[6] = {p0, p1, p2, p3, inp1v, inp6v};
    float prevA[4] = {p0, p1, p2, p3};
    float d[4];
#pragma unroll
    for (int i = 0; i < 4; ++i) {
      float s = 0.0f;
#pragma unroll
      for (int j = 0; j < 6; ++j)
        s += (temps[j] - prevA[i]) * cond[ADJ[i][j]];
      d[i] = s;
    }

    p0 = clip15(p0 + k0 * (d[0] + fabsf(pl4.x)));
    p1 = clip15(p1 + k1 * (d[1] + fabsf(pl4.y)));
    p2 = clip15(p2 + k2 * (d[2] + fabsf(pl4.z)));
    p3 = clip15(p3 + k3 * (d[3] + fabsf(pl4.w)));

    A0 = A0n; A1 = A1n;
    toff  += (size_t)NB * 8;
    otoff += (size_t)NB * 4;
  }

  // final_state appended after outputs (same half-tiled b64 pattern)
  {
    v2f fin;
    fin.x = hi ? p2 : p0;
    fin.y = hi ? p3 : p1;
    *(v2f*)(out + (size_t)NT * NB * 4 + (size_t)b * 4 + half * 2) = fin;
  }
}

extern "C" void kernel_launch(void* const* d_in, const int* in_sizes, int n_in,
                              void* d_out, int out_size, void* d_ws, size_t ws_size,
                              hipStream_t stream) {
  const float* inp   = (const float*)d_in[0];
  const float* state = (const float*)d_in[1];
  const float* caps  = (const float*)d_in[2];
  const float* Wc    = (const float*)d_in[3];
  const float* bc    = (const float*)d_in[4];
  const float* W1    = (const float*)d_in[5];
  const float* b1    = (const float*)d_in[6];
  const float* W2    = (const float*)d_in[7];
  const float* b2    = (const float*)d_in[8];
  // d_in[9] = adj_mat: compile-time constant, unused at runtime
  float* out = (float*)d_out;

  dim3 grid(NB / 128);  // 16 blocks; 8 waves/block, 16 batches/wave
  dim3 block(256);
  hipLaunchKernelGGL(diffeq_wmma_kernel, grid, block, 0, stream,
                     inp, state, caps, Wc, bc, W1, b1, W2, b2, out);
}